// ManifoldnessConstraint_20710332301899
// MI455X (gfx1250) — compile-verified
//
#include <hip/hip_runtime.h>
#include <hip/hip_bf16.h>
#include <math.h>

#define BATCH   4
#define NPTS    4096
#define NEIGH   32
#define SUPPORT 8
#define EPSV    1e-6f
#define NTILES  (NPTS / 16)

typedef __attribute__((ext_vector_type(2))) float v2f;
typedef __attribute__((ext_vector_type(8))) float v8f;
typedef __attribute__((ext_vector_type(2))) int   v2i;

// ---- CDNA5 async global->LDS path (ASYNCcnt-tracked), guarded so a missing
// ---- builtin degrades to synchronous LDS staging instead of a compile error.
#if defined(__has_builtin)
#  if __has_builtin(__builtin_amdgcn_global_load_async_to_lds_b64) && \
      __has_builtin(__builtin_amdgcn_s_wait_asynccnt)
#    define USE_ASYNC_LDS 1
#  endif
#endif

#ifdef USE_ASYNC_LDS
// b64 variant is typed as pointer-to-int2: AS(1) source, AS(3) destination.
typedef __attribute__((address_space(1))) v2i* gptr_v2i;
typedef __attribute__((address_space(3))) v2i* lptr_v2i;
__device__ __forceinline__ void async_copy_b64(const float* gsrc, float* ldst) {
    // global_load_async_to_lds_b64: 8 bytes per lane, ASYNCcnt-tracked
    __builtin_amdgcn_global_load_async_to_lds_b64(
        (gptr_v2i)(uintptr_t)gsrc,
        (lptr_v2i)(unsigned int)(uintptr_t)ldst,
        0, 0);
}
#endif

// ---------------------------------------------------------------------------
// Kernel 1: k-NN search. One wave32 per 16 query points.
// Distance tile d[16x16] = |q|^2 + |c|^2 - 2 q.c ; the Gram part is computed
// with V_WMMA_F32_16X16X4_F32 (K=4 holds xyz padded with 0; A pre-scaled by -2).
// Candidate tiles (16 pts * 12 B = 192 B) are double-buffered in LDS via
// async global->LDS copies so the fetch of tile t+1 overlaps WMMA + top-32
// selection on tile t.  Top-32 selection via per-query insertion (lanes 0..15).
// ---------------------------------------------------------------------------
__global__ __launch_bounds__(32)
void knn_wmma_kernel(const float* __restrict__ xyz, int* __restrict__ idx_out) {
    __shared__ float sqq[16];
    __shared__ float cbuf[2][48];  // double-buffered candidate tile (16 x xyz)
    __shared__ float dt[16][17];   // distance tile, padded vs bank conflicts

    const int lane = threadIdx.x;          // 0..31
    const int b    = blockIdx.x / NTILES;
    const int q0   = (blockIdx.x % NTILES) * 16;
    const float* P = xyz + (size_t)b * NPTS * 3;

    const int m    = lane & 15;
    const int half = lane >> 4;            // 0: K=0,1   1: K=2,3
    const int n    = lane & 15;            // candidate column owned by lane

    // --- A fragment: 16x4 (M x K), row m striped across VGPRs in lane m / m+16.
    // Fold in the -2 factor so WMMA directly yields -2 * <q,c>.
    const float qx = P[(q0 + m) * 3 + 0];
    const float qy = P[(q0 + m) * 3 + 1];
    const float qz = P[(q0 + m) * 3 + 2];
    v2f afrag;
    afrag.x = half ? (-2.0f * qz) : (-2.0f * qx);   // K=2 : K=0
    afrag.y = half ? 0.0f         : (-2.0f * qy);   // K=3 : K=1
    if (lane < 16) sqq[lane] = qx * qx + qy * qy + qz * qz;

    // --- per-query top-32 (ascending) lists, lanes 0..15 own one query each
    float bestd[NEIGH];
    int   besti[NEIGH];
#pragma unroll
    for (int k = 0; k < NEIGH; ++k) { bestd[k] = 3.4e38f; besti[k] = 0; }

    // --- stage tile 0 (lanes 0..23 move 8 B each = 192 B)
#ifdef USE_ASYNC_LDS
    if (lane < 24) async_copy_b64(P + lane * 2, &cbuf[0][lane * 2]);
#else
    if (lane < 24) {
        const float2 v = *reinterpret_cast<const float2*>(P + lane * 2);
        cbuf[0][lane * 2 + 0] = v.x;
        cbuf[0][lane * 2 + 1] = v.y;
    }
#endif

    for (int t = 0; t < NTILES; ++t) {
        const int buf = t & 1;

        // issue the copy for tile t+1, then wait for tile t (in-order completion)
        if (t + 1 < NTILES) {
            const float* src = P + (size_t)(t + 1) * 48;
#ifdef USE_ASYNC_LDS
            if (lane < 24) async_copy_b64(src + lane * 2, &cbuf[buf ^ 1][lane * 2]);
            __builtin_amdgcn_s_wait_asynccnt(1);
#else
            if (lane < 24) {
                __builtin_prefetch(src + lane * 2, 0, 0);   // global_prefetch_b8
                const float2 v = *reinterpret_cast<const float2*>(src + lane * 2);
                cbuf[(buf ^ 1)][lane * 2 + 0] = v.x;
                cbuf[(buf ^ 1)][lane * 2 + 1] = v.y;
            }
#endif
        }
#ifdef USE_ASYNC_LDS
        else {
            __builtin_amdgcn_s_wait_asynccnt(0);
        }
#endif
        __syncthreads();   // tile t visible to all lanes; prior dt consumed

        // --- each lane owns candidate n of this tile (read from LDS)
        const float cx = cbuf[buf][n * 3 + 0];
        const float cy = cbuf[buf][n * 3 + 1];
        const float cz = cbuf[buf][n * 3 + 2];
        const float sqc = cx * cx + cy * cy + cz * cz;

        // --- B fragment: 4x16 (K x N), row K striped across lanes per VGPR:
        // v0: lanes0-15 = K0(x), lanes16-31 = K1(y); v1: K2(z), K3(0)
        v2f bfrag;
        bfrag.x = half ? cy   : cx;
        bfrag.y = half ? 0.0f : cz;

        v8f acc = {};
        // 8 args: (neg_a, A, neg_b, B, c_mod, C, reuse_a, reuse_b)
        acc = __builtin_amdgcn_wmma_f32_16x16x4_f32(
                  false, afrag, false, bfrag, (short)0, acc, false, false);

        // C/D layout: VGPR r holds M = r + 8*half, N = lane%16
#pragma unroll
        for (int r = 0; r < 8; ++r) {
            const int mr = r + 8 * half;
            dt[mr][n] = sqq[mr] + sqc + acc[r];
        }
        __syncthreads();

        if (lane < 16) {
            const int c0 = t * 16;
#pragma unroll 1
            for (int j = 0; j < 16; ++j) {
                const float d = dt[lane][j];
                if (d < bestd[NEIGH - 1]) {
                    int pos = NEIGH - 1;
                    while (pos > 0 && bestd[pos - 1] > d) {
                        bestd[pos] = bestd[pos - 1];
                        besti[pos] = besti[pos - 1];
                        --pos;
                    }
                    bestd[pos] = d;
                    besti[pos] = c0 + j;
                }
            }
        }
        __syncthreads();
    }

    if (lane < 16) {
        int* dst = idx_out + ((size_t)b * NPTS + q0 + lane) * NEIGH;
#pragma unroll 1
        for (int k = 0; k < NEIGH; ++k) dst[k] = besti[k];
    }
}

// ---------------------------------------------------------------------------
// Kernel 2: per-point covariance + analytic smallest eigenvector + orientation
// ---------------------------------------------------------------------------
__global__ void normals_kernel(const float* __restrict__ xyz,
                               const int* __restrict__ idx_ws,
                               float* __restrict__ nrm_ws) {
    const int t = blockIdx.x * blockDim.x + threadIdx.x;
    if (t >= BATCH * NPTS) return;
    const int b = t / NPTS;
    const int n = t % NPTS;
    const float* P  = xyz + (size_t)b * NPTS * 3;
    const int*   nb = idx_ws + (size_t)t * NEIGH;

    float px[NEIGH], py[NEIGH], pz[NEIGH];
    float mx = 0.f, my = 0.f, mz = 0.f;
#pragma unroll 1
    for (int k = 0; k < NEIGH; ++k) {
        const int j = nb[k];
        const float x = P[j * 3 + 0], y = P[j * 3 + 1], z = P[j * 3 + 2];
        px[k] = x; py[k] = y; pz[k] = z;
        mx += x; my += y; mz += z;
    }
    mx *= (1.0f / NEIGH); my *= (1.0f / NEIGH); mz *= (1.0f / NEIGH);

    float cxx = 0.f, cxy = 0.f, cxz = 0.f, cyy = 0.f, cyz = 0.f, czz = 0.f;
#pragma unroll 1
    for (int k = 0; k < NEIGH; ++k) {
        const float x = px[k] - mx, y = py[k] - my, z = pz[k] - mz;
        cxx += x * x; cxy += x * y; cxz += x * z;
        cyy += y * y; cyz += y * z; czz += z * z;
    }

    // Smallest eigenvalue of symmetric 3x3 (trigonometric method)
    const float p1 = cxy * cxy + cxz * cxz + cyz * cyz;
    const float qd = (cxx + cyy + czz) * (1.0f / 3.0f);
    const float p2 = (cxx - qd) * (cxx - qd) + (cyy - qd) * (cyy - qd) +
                     (czz - qd) * (czz - qd) + 2.0f * p1;
    const float p  = sqrtf(fmaxf(p2, 0.0f) * (1.0f / 6.0f));
    float lam = qd;
    if (p > 1e-20f) {
        const float ib  = 1.0f / p;
        const float b00 = (cxx - qd) * ib, b01 = cxy * ib, b02 = cxz * ib;
        const float b11 = (cyy - qd) * ib, b12 = cyz * ib, b22 = (czz - qd) * ib;
        float r = 0.5f * (b00 * (b11 * b22 - b12 * b12)
                        - b01 * (b01 * b22 - b12 * b02)
                        + b02 * (b01 * b12 - b11 * b02));
        r = fminf(1.0f, fmaxf(-1.0f, r));
        const float phi = acosf(r) * (1.0f / 3.0f);
        lam = qd + 2.0f * p * cosf(phi + 2.0943951023931953f);  // smallest
    }

    // Eigenvector via best cross-product of rows of (A - lam*I)
    const float r0x = cxx - lam, r0y = cxy,       r0z = cxz;
    const float r1x = cxy,       r1y = cyy - lam, r1z = cyz;
    const float r2x = cxz,       r2y = cyz,       r2z = czz - lam;
    float c0x = r0y * r1z - r0z * r1y, c0y = r0z * r1x - r0x * r1z, c0z = r0x * r1y - r0y * r1x;
    float c1x = r0y * r2z - r0z * r2y, c1y = r0z * r2x - r0x * r2z, c1z = r0x * r2y - r0y * r2x;
    float c2x = r1y * r2z - r1z * r2y, c2y = r1z * r2x - r1x * r2z, c2z = r1x * r2y - r1y * r2x;
    const float l0 = c0x * c0x + c0y * c0y + c0z * c0z;
    const float l1 = c1x * c1x + c1y * c1y + c1z * c1z;
    const float l2 = c2x * c2x + c2y * c2y + c2z * c2z;
    float vx, vy, vz, ln;
    if (l0 >= l1 && l0 >= l2)      { vx = c0x; vy = c0y; vz = c0z; ln = l0; }
    else if (l1 >= l2)             { vx = c1x; vy = c1y; vz = c1z; ln = l1; }
    else                           { vx = c2x; vy = c2y; vz = c2z; ln = l2; }
    if (ln < 1e-30f) { vx = 1.0f; vy = 0.0f; vz = 0.0f; ln = 1.0f; }
    const float inv = rsqrtf(ln);
    vx *= inv; vy *= inv; vz *= inv;

    // Orientation: flip if sum_k <n, nbr_k - x> < 0
    const float ox = P[n * 3 + 0], oy = P[n * 3 + 1], oz = P[n * 3 + 2];
    float proj = 0.f;
#pragma unroll 1
    for (int k = 0; k < NEIGH; ++k)
        proj += vx * (px[k] - ox) + vy * (py[k] - oy) + vz * (pz[k] - oz);
    if (proj < 0.0f) { vx = -vx; vy = -vy; vz = -vz; }

    nrm_ws[(size_t)t * 3 + 0] = vx;
    nrm_ws[(size_t)t * 3 + 1] = vy;
    nrm_ws[(size_t)t * 3 + 2] = vz;
}

// ---------------------------------------------------------------------------
// Kernel 3: per-point penalty std over SUPPORT neighbors (ddof=1)
// ---------------------------------------------------------------------------
__global__ void penalty_kernel(const int* __restrict__ idx_ws,
                               const float* __restrict__ nrm_ws,
                               float* __restrict__ pen_ws) {
    const int t = blockIdx.x * blockDim.x + threadIdx.x;
    if (t >= BATCH * NPTS) return;
    const int b = t / NPTS;
    const int* nb = idx_ws + (size_t)t * NEIGH;
    const float* NB = nrm_ws + (size_t)b * NPTS * 3;

    const int j0 = nb[0];
    const float ax = NB[j0 * 3 + 0], ay = NB[j0 * 3 + 1], az = NB[j0 * 3 + 2];
    const float na = fmaxf(sqrtf(ax * ax + ay * ay + az * az), EPSV);

    float pen[SUPPORT];
    float mean = 0.f;
#pragma unroll
    for (int k = 0; k < SUPPORT; ++k) {
        const int j = nb[k];
        const float nx = NB[j * 3 + 0], ny = NB[j * 3 + 1], nz = NB[j * 3 + 2];
        const float nn = fmaxf(sqrtf(nx * nx + ny * ny + nz * nz), EPSV);
        const float c  = (ax * nx + ay * ny + az * nz) / (na * nn);
        pen[k] = 1.0f - c;
        mean  += pen[k];
    }
    mean *= (1.0f / SUPPORT);
    float var = 0.f;
#pragma unroll
    for (int k = 0; k < SUPPORT; ++k) {
        const float d = pen[k] - mean;
        var += d * d;
    }
    pen_ws[t] = sqrtf(var * (1.0f / (SUPPORT - 1)));
}

// ---------------------------------------------------------------------------
// Kernel 4: deterministic per-batch mean (fixed-order strided sum + LDS tree)
// ---------------------------------------------------------------------------
__global__ void reduce_kernel(const float* __restrict__ pen_ws,
                              float* __restrict__ out) {
    __shared__ float s[256];
    const int b = blockIdx.x;
    float acc = 0.f;
    for (int i = threadIdx.x; i < NPTS; i += 256)
        acc += pen_ws[(size_t)b * NPTS + i];
    s[threadIdx.x] = acc;
    __syncthreads();
    for (int st = 128; st > 0; st >>= 1) {
        if (threadIdx.x < st) s[threadIdx.x] += s[threadIdx.x + st];
        __syncthreads();
    }
    if (threadIdx.x == 0) out[b] = s[0] * (1.0f / NPTS);
}

// ---------------------------------------------------------------------------
extern "C" void kernel_launch(void* const* d_in, const int* in_sizes, int n_in,
                              void* d_out, int out_size, void* d_ws, size_t ws_size,
                              hipStream_t stream) {
    (void)in_sizes; (void)n_in; (void)out_size; (void)ws_size;
    const float* xyz = (const float*)d_in[0];
    float* out = (float*)d_out;

    // workspace layout: idx[B*N*32] (int) | normals[B*N*3] (f32) | penalty[B*N] (f32)
    int*   idx_ws = (int*)d_ws;
    float* nrm_ws = (float*)((char*)d_ws + (size_t)BATCH * NPTS * NEIGH * sizeof(int));
    float* pen_ws = nrm_ws + (size_t)BATCH * NPTS * 3;

    knn_wmma_kernel<<<BATCH * NTILES, 32, 0, stream>>>(xyz, idx_ws);
    normals_kernel<<<(BATCH * NPTS + 255) / 256, 256, 0, stream>>>(xyz, idx_ws, nrm_ws);
    penalty_kernel<<<(BATCH * NPTS + 255) / 256, 256, 0, stream>>>(idx_ws, nrm_ws, pen_ws);
    reduce_kernel<<<BATCH, 256, 0, stream>>>(pen_ws, out);
}